// Gate_Net_86268713107863
// MI455X (gfx1250) — compile-verified
//
#include <hip/hip_runtime.h>
#include <stdint.h>

// Problem geometry (fixed by the reference: B=32, L=1024).
#define LDIM   1024
#define NCOL   1022          // n   = L - 2   (columns j)
#define NROW   1021          // n-1           (cumprod rows k)
#define PADZ   1022          // leading zeros so t[j-1-k] for j-1-k<0 reads 0 branch-free
#define TPB    128           // 4 waves (wave32)
#define CB     4             // column-pair sub-blocks per (batch, direction)

typedef float v2f __attribute__((ext_vector_type(2)));

// out[b][dir][k][j] = prod_{m<=k} prob(t[j-1-m] (0 if neg idx), t[j])
//   where t = docs[b][1:-1] (dir=0) or reversed (dir=1),
//   prob(a,b) = 0.5*(clamp((a-b)*20+1, -1, 1)+1)
__global__ __launch_bounds__(TPB) void gate_kernel(
        const float* __restrict__ score,
        const int*   __restrict__ score_idx,
        float*       __restrict__ out)
{
    __shared__ float s_pad[PADZ + NCOL];   // [0,PADZ): zeros, [PADZ,...): t[0..NCOL)

    const int blk = blockIdx.x;
    const int cb  = blk & (CB - 1);        // column-pair sub-block
    const int dir = (blk >> 2) & 1;        // 0 = fwd, 1 = bwd
    const int b   = blk >> 3;              // batch row

    // Zero the left pad (covers all j-1-k < 0 accesses).
    for (int i = threadIdx.x; i < PADZ; i += TPB) s_pad[i] = 0.0f;

    // Stage the (possibly reversed) document slice into LDS with the CDNA5
    // async-to-LDS path: per-lane gather through score_idx, no VGPR round-trip.
    for (int m = threadIdx.x; m < NCOL; m += TPB) {
        const int l = dir ? (LDIM - 2 - m) : (m + 1);
        const int g = score_idx[b * LDIM + l];
        const uint32_t lds_off = (uint32_t)(uintptr_t)(&s_pad[PADZ + m]); // low 32b = LDS offset
        const uint64_t gaddr   = (uint64_t)(uintptr_t)(score + g);
        asm volatile("global_load_async_to_lds_b32 %0, %1, off"
                     :: "v"(lds_off), "v"(gaddr) : "memory");
    }
    asm volatile("s_wait_asynccnt 0" ::: "memory");
    __syncthreads();

    const int pair = cb * TPB + threadIdx.x;   // pair index, columns (2p, 2p+1)
    const int j0   = pair * 2;
    if (j0 < NCOL) {
        const float b0 = s_pad[PADZ + j0];
        const float b1 = s_pad[PADZ + j0 + 1];
        // z = (a-b)*20+1 = fma(a, 20, c),  c = fma(-20, b, 1)
        const float c0 = __builtin_fmaf(-20.0f, b0, 1.0f);
        const float c1 = __builtin_fmaf(-20.0f, b1, 1.0f);

        float a_hi = b0;                       // t[j0   - k] at k=0
        float a_lo = s_pad[PADZ + j0 - 1];     // t[j0-1 - k] at k=0
        float acc0 = 1.0f, acc1 = 1.0f;

        const size_t base = ((size_t)(b * 2 + dir) * NROW) * NCOL + (size_t)j0;
        v2f* op = (v2f*)(out + base);          // even element offset -> 8B aligned

        #pragma unroll 4
        for (int k = 0; k < NROW; ++k) {
            float z0 = __builtin_fmaf(a_lo, 20.0f, c0);
            float z1 = __builtin_fmaf(a_hi, 20.0f, c1);
            z0 = fminf(fmaxf(z0, -1.0f), 1.0f);
            z1 = fminf(fmaxf(z1, -1.0f), 1.0f);
            acc0 *= __builtin_fmaf(z0, 0.5f, 0.5f);
            acc1 *= __builtin_fmaf(z1, 0.5f, 0.5f);
            v2f v; v.x = acc0; v.y = acc1;
            __builtin_nontemporal_store(v, op);   // 267 MB stream-out: keep L2 clean
            op += NCOL / 2;
            // slide the window: next row needs t[j0-2-k] and previous a_lo
            a_hi = a_lo;
            a_lo = s_pad[PADZ + j0 - 2 - k];      // min index: PADZ - 1022 == 0
        }
    }
}

extern "C" void kernel_launch(void* const* d_in, const int* in_sizes, int n_in,
                              void* d_out, int out_size, void* d_ws, size_t ws_size,
                              hipStream_t stream) {
    const float* score     = (const float*)d_in[0];
    const int*   score_idx = (const int*)d_in[1];
    float*       out       = (float*)d_out;

    const int Bn = in_sizes[1] / LDIM;            // 32
    const dim3 grid(Bn * 2 * CB);                 // 256 blocks x 128 threads = 1024 waves
    gate_kernel<<<grid, TPB, 0, stream>>>(score, score_idx, out);
}